// Loss_60430189855003
// MI455X (gfx1250) — compile-verified
//
#include <hip/hip_runtime.h>
#include <math.h>

// ---------------- problem constants ----------------
constexpr int kB  = 32;
constexpr int kN  = 1024;
constexpr int kBN = kB * kN;

// ---------------- workspace layout (floats) ----------------
constexpr size_t O_ROWMAX1 = 0;
constexpr size_t O_ROWMAX2 = (size_t)1 * kBN;
constexpr size_t O_COLMAX1 = (size_t)2 * kBN;
constexpr size_t O_COLMAX2 = (size_t)3 * kBN;
constexpr size_t O_COLSUM1 = (size_t)4 * kBN;
constexpr size_t O_COLSUM2 = (size_t)5 * kBN;
constexpr size_t O_CS1_1   = (size_t)6 * kBN;   // sum_n sa1        per column
constexpr size_t O_CS2_1   = (size_t)7 * kBN;   // sum_n sa1*a1     per column
constexpr size_t O_CS1_2   = (size_t)8 * kBN;
constexpr size_t O_CS2_2   = (size_t)9 * kBN;
constexpr size_t O_ROWENT1 = (size_t)10 * kBN;
constexpr size_t O_ROWENT2 = (size_t)11 * kBN;
constexpr size_t O_RMIN0   = (size_t)12 * kBN;  // chamfer1 row mins
constexpr size_t O_CMIN0   = (size_t)13 * kBN;  // chamfer1 col mins
constexpr size_t O_RMIN1   = (size_t)14 * kBN;
constexpr size_t O_CMIN1   = (size_t)15 * kBN;
constexpr size_t O_P1IN2   = (size_t)16 * kBN;  // 3*kBN
constexpr size_t O_P2IN1   = (size_t)19 * kBN;  // 3*kBN
constexpr size_t O_COS     = (size_t)22 * kBN;  // dot[32], s1[32], s2[32]
constexpr size_t O_POSE    = (size_t)22 * kBN + 96; // 32 * 24 floats
constexpr size_t WS_FLOATS = O_POSE + (size_t)24 * kB;

typedef float v2f __attribute__((ext_vector_type(2)));
typedef float v8f __attribute__((ext_vector_type(8)));

// ---------------- wave32 helpers ----------------
__device__ inline float waveSum(float v) {
#pragma unroll
    for (int o = 16; o > 0; o >>= 1) v += __shfl_xor(v, o, 32);
    return v;
}
__device__ inline float waveMaxF(float v) {
#pragma unroll
    for (int o = 16; o > 0; o >>= 1) v = fmaxf(v, __shfl_xor(v, o, 32));
    return v;
}
__device__ inline void atomicMaxF(float* a, float v) {
    int cur = __float_as_int(*a);
    while (v > __int_as_float(cur)) {
        int old = atomicCAS((int*)a, cur, __float_as_int(v));
        if (old == cur) break;
        cur = old;
    }
}

// ---------------- init workspace ----------------
__global__ void k_init(float* ws) {
    size_t i = (size_t)blockIdx.x * blockDim.x + threadIdx.x;
    if (i >= WS_FLOATS) return;
    if (i < O_COLSUM1) {                      // row/col maxima
        ws[i] = -__builtin_inff();
    } else if (i < O_ROWENT1) {               // colsum / S1 / S2 accumulators
        ws[i] = 0.0f;
    } else if (i >= O_RMIN0 && i < O_P1IN2) { // chamfer mins
        ws[i] = __builtin_inff();
    } else if (i >= O_COS && i < O_COS + 96) {
        ws[i] = 0.0f;
    }
}

// ---------------- pose: sRt and analytic inverse ----------------
__global__ void k_pose(const float* __restrict__ scale,
                       const float* __restrict__ rot,
                       const float* __restrict__ trans, float* ws) {
    int b = threadIdx.x;
    if (b >= kB) return;
    float s = scale[b];
    float A[9];
#pragma unroll
    for (int i = 0; i < 9; ++i) A[i] = s * rot[b * 9 + i];
    float t0 = trans[b * 3], t1 = trans[b * 3 + 1], t2 = trans[b * 3 + 2];
    float det = A[0] * (A[4] * A[8] - A[5] * A[7])
              - A[1] * (A[3] * A[8] - A[5] * A[6])
              + A[2] * (A[3] * A[7] - A[4] * A[6]);
    float id = 1.0f / det;
    float I[9];
    I[0] = (A[4] * A[8] - A[5] * A[7]) * id;
    I[1] = (A[2] * A[7] - A[1] * A[8]) * id;
    I[2] = (A[1] * A[5] - A[2] * A[4]) * id;
    I[3] = (A[5] * A[6] - A[3] * A[8]) * id;
    I[4] = (A[0] * A[8] - A[2] * A[6]) * id;
    I[5] = (A[2] * A[3] - A[0] * A[5]) * id;
    I[6] = (A[3] * A[7] - A[4] * A[6]) * id;
    I[7] = (A[1] * A[6] - A[0] * A[7]) * id;
    I[8] = (A[0] * A[4] - A[1] * A[3]) * id;
    float u0 = -(I[0] * t0 + I[1] * t1 + I[2] * t2);
    float u1 = -(I[3] * t0 + I[4] * t1 + I[5] * t2);
    float u2 = -(I[6] * t0 + I[7] * t1 + I[8] * t2);
    float* p = ws + O_POSE + b * 24;
#pragma unroll
    for (int i = 0; i < 9; ++i) p[i] = A[i];
    p[9] = t0; p[10] = t1; p[11] = t2;
#pragma unroll
    for (int i = 0; i < 9; ++i) p[12 + i] = I[i];
    p[21] = u0; p[22] = u1; p[23] = u2;
}

// ---------------- pass 1: row/col maxima + cosine sums ----------------
// grid (B, 4 colstrips, 4 rowstrips), block 256
__global__ void k_pass1(const float* __restrict__ A1,
                        const float* __restrict__ A2, float* ws) {
    int b = blockIdx.x, cs = blockIdx.y, rs = blockIdx.z;
    int t = threadIdx.x, lane = t & 31, w = t >> 5;
    int c = cs * 256 + t;
    const float* a1 = A1 + (size_t)b * kN * kN + c;
    const float* a2 = A2 + (size_t)b * kN * kN + c;
    float cm1 = -__builtin_inff(), cm2 = -__builtin_inff();
    float dot = 0.f, s1 = 0.f, s2 = 0.f;
    int n0 = rs * 256;
    for (int k = 0; k < 256; ++k) {
        int n = n0 + k;
        __builtin_prefetch(a1 + (size_t)(n + 8) * kN, 0, 0);
        float x1 = a1[(size_t)n * kN];
        float x2 = a2[(size_t)n * kN];
        cm1 = fmaxf(cm1, x1);
        cm2 = fmaxf(cm2, x2);
        dot = fmaf(x1, x2, dot);
        s1  = fmaf(x1, x1, s1);
        s2  = fmaf(x2, x2, s2);
        float m1 = waveMaxF(x1);
        float m2 = waveMaxF(x2);
        if (lane == 0) {
            atomicMaxF(ws + O_ROWMAX1 + b * kN + n, m1);
            atomicMaxF(ws + O_ROWMAX2 + b * kN + n, m2);
        }
    }
    atomicMaxF(ws + O_COLMAX1 + b * kN + c, cm1);
    atomicMaxF(ws + O_COLMAX2 + b * kN + c, cm2);
    __shared__ float red[3][8];
    dot = waveSum(dot); s1 = waveSum(s1); s2 = waveSum(s2);
    if (lane == 0) { red[0][w] = dot; red[1][w] = s1; red[2][w] = s2; }
    __syncthreads();
    if (t == 0) {
        float d = 0, e = 0, f = 0;
        for (int i = 0; i < 8; ++i) { d += red[0][i]; e += red[1][i]; f += red[2][i]; }
        atomicAdd(ws + O_COS + b, d);
        atomicAdd(ws + O_COS + 32 + b, e);
        atomicAdd(ws + O_COS + 64 + b, f);
    }
}

// ---------------- pass 2: softmax streaming (row stats + column partials) ----------------
// grid (B, 128 row-blocks of 8), block 256, one wave per row
__global__ void k_softmax(const float* __restrict__ A,
                          const float* __restrict__ Pother,
                          float* ws, int which) {
    int b = blockIdx.x, rb = blockIdx.y;
    int t = threadIdx.x, lane = t & 31, w = t >> 5;
    const float* rowmax_g = ws + (which ? O_ROWMAX2 : O_ROWMAX1) + b * kN;
    const float* colmax_g = ws + (which ? O_COLMAX2 : O_COLMAX1) + b * kN;
    float* colsum_g = ws + (which ? O_COLSUM2 : O_COLSUM1) + b * kN;
    float* cs1_g    = ws + (which ? O_CS1_2 : O_CS1_1) + b * kN;
    float* cs2_g    = ws + (which ? O_CS2_2 : O_CS2_1) + b * kN;
    float* rowent_g = ws + (which ? O_ROWENT2 : O_ROWENT1) + b * kN;
    float* pin_g    = ws + (which ? O_P2IN1 : O_P1IN2) + (size_t)b * kN * 3;

    __shared__ float pts[3 * kN];
    __shared__ float cmx[kN];
    __shared__ float csum[kN];
    __shared__ float cS1[kN];
    __shared__ float cS2[kN];
    for (int k = t; k < 3 * kN; k += 256) pts[k] = Pother[(size_t)b * 3 * kN + k];
    for (int k = t; k < kN; k += 256) {
        cmx[k] = colmax_g[k]; csum[k] = 0.f; cS1[k] = 0.f; cS2[k] = 0.f;
    }
    __syncthreads();

    int n = rb * 8 + w;
    const float* arow = A + ((size_t)b * kN + n) * kN;
    float rmax = rowmax_g[n];
    float se = 0.f, sea = 0.f, wp0 = 0.f, wp1 = 0.f, wp2 = 0.f;
#pragma unroll 4
    for (int i = 0; i < 32; ++i) {
        int c = (((i + w * 4) & 31) * 32) + lane;  // stagger waves across banks
        float a = arow[c];
        float e = __expf(a - rmax);
        se += e;
        sea = fmaf(e, a, sea);
        wp0 = fmaf(e, pts[3 * c + 0], wp0);
        wp1 = fmaf(e, pts[3 * c + 1], wp1);
        wp2 = fmaf(e, pts[3 * c + 2], wp2);
        atomicAdd(&csum[c], __expf(a - cmx[c]));
    }
    se  = waveSum(se);
    sea = waveSum(sea);
    wp0 = waveSum(wp0); wp1 = waveSum(wp1); wp2 = waveSum(wp2);
    float inv = 1.0f / se;
    if (lane == 0) {
        rowent_g[n] = -(sea * inv) + rmax + __logf(se);   // -sum sa*lsm2 (sum sa == 1)
        pin_g[3 * n + 0] = wp0 * inv;
        pin_g[3 * n + 1] = wp1 * inv;
        pin_g[3 * n + 2] = wp2 * inv;
    }
    // second sweep (strip is L1/L2 hot): column sums of sa and sa*a
#pragma unroll 4
    for (int i = 0; i < 32; ++i) {
        int c = (((i + w * 4) & 31) * 32) + lane;
        float a = arow[c];
        float e = __expf(a - rmax) * inv;
        atomicAdd(&cS1[c], e);
        atomicAdd(&cS2[c], e * a);
    }
    __syncthreads();
    for (int k = t; k < kN; k += 256) {
        atomicAdd(&colsum_g[k], csum[k]);
        atomicAdd(&cs1_g[k], cS1[k]);
        atomicAdd(&cs2_g[k], cS2[k]);
    }
}

// ---------------- chamfer via V_WMMA_F32_16X16X4_F32 ----------------
// grid (B, 2 pairs, 8 rowstrips), block 256 (8 waves, 16 rows per wave)
__global__ void k_chamfer(const float* __restrict__ P1,
                          const float* __restrict__ P2, float* ws) {
    int b = blockIdx.x, pair = blockIdx.y, rstrip = blockIdx.z;
    int t = threadIdx.x, lane = t & 31, w = t >> 5;
    int half = lane >> 4, l15 = lane & 15;
    const float* pose = ws + O_POSE + b * 24 + pair * 12; // [A,t] or [Ainv,t2]
    const float* P = pair ? P2 : P1;
    const float* Y = ws + (pair ? O_P2IN1 : O_P1IN2) + (size_t)b * kN * 3;
    float* rmin = ws + (pair ? O_RMIN1 : O_RMIN0) + b * kN;
    float* cmin = ws + (pair ? O_CMIN1 : O_CMIN0) + b * kN;

    // stage the 1024 y-points + |y|^2 in LDS (reused 64x by every wave)
    __shared__ float ylds[3 * kN];
    __shared__ float yy[kN];
    for (int k = t; k < 3 * kN; k += 256) ylds[k] = Y[k];
    __syncthreads();
    for (int k = t; k < kN; k += 256) {
        float y0 = ylds[3 * k], y1 = ylds[3 * k + 1], y2 = ylds[3 * k + 2];
        yy[k] = y0 * y0 + y1 * y1 + y2 * y2;
    }

    float M0 = pose[0], M1 = pose[1], M2 = pose[2];
    float M3 = pose[3], M4 = pose[4], M5 = pose[5];
    float M6 = pose[6], M7 = pose[7], M8 = pose[8];
    float T0 = pose[9], T1 = pose[10], T2 = pose[11];

    int rowbase = rstrip * 128 + w * 16;
    int myrow = rowbase + l15;
    const float* pp = P + ((size_t)b * kN + myrow) * 3;
    float p0 = pp[0], p1 = pp[1], p2 = pp[2];
    float tx = fmaf(M0, p0, fmaf(M1, p1, fmaf(M2, p2, T0)));
    float ty = fmaf(M3, p0, fmaf(M4, p1, fmaf(M5, p2, T1)));
    float tz = fmaf(M6, p0, fmaf(M7, p1, fmaf(M8, p2, T2)));
    float xx2 = tx * tx + ty * ty + tz * tz;

    // A fragment 16x4 f32: lanes0-15 hold K=0,1 ; lanes16-31 hold K=2,3 (pad 0)
    v2f afrag;
    afrag.x = half ? tz : tx;
    afrag.y = half ? 0.0f : ty;

    // hoist the row-|x|^2 broadcasts out of the tile loop (loop-invariant):
    // lane needs |x|^2 of rows M = v + 8*half, held by lane (v + 8*half)
    float xm[8];
#pragma unroll
    for (int v = 0; v < 8; ++v) xm[v] = __shfl(xx2, v + (half << 3), 32);

    float racc[8];
#pragma unroll
    for (int v = 0; v < 8; ++v) racc[v] = __builtin_inff();

    __syncthreads();  // yy ready

    for (int mt = 0; mt < 64; ++mt) {
        int m = mt * 16 + l15;
        float y0 = ylds[3 * m], y1 = ylds[3 * m + 1], y2 = ylds[3 * m + 2];
        float yy2 = yy[m];
        v2f bfrag;
        bfrag.x = half ? y2 : y0;
        bfrag.y = half ? 0.0f : y1;
        v8f cz = {0.f, 0.f, 0.f, 0.f, 0.f, 0.f, 0.f, 0.f};
        v8f c = __builtin_amdgcn_wmma_f32_16x16x4_f32(
            false, afrag, false, bfrag, (short)0, cz, false, false);
        float cmv = __builtin_inff();
#pragma unroll
        for (int v = 0; v < 8; ++v) {
            // C layout: row M = v + 8*half, col N = lane&15
            float d = fmaxf(fmaf(-2.0f, c[v], xm[v] + yy2), 0.0f);
            racc[v] = fminf(racc[v], d);
            cmv = fminf(cmv, d);
        }
        // combine the two half-wave partials, one atomic per column
        cmv = fminf(cmv, __shfl_xor(cmv, 16, 32));
        if (half == 0)
            atomicMin((int*)&cmin[m], __float_as_int(cmv)); // d >= 0: int order ok
    }
#pragma unroll
    for (int v = 0; v < 8; ++v) {
#pragma unroll
        for (int o = 1; o < 16; o <<= 1)
            racc[v] = fminf(racc[v], __shfl_xor(racc[v], o, 32));
    }
    if (l15 == 0) {
#pragma unroll
        for (int v = 0; v < 8; ++v)
            rmin[rowbase + v + (half << 3)] = racc[v];
    }
}

// ---------------- final scalar reduction ----------------
__global__ void k_final(const float* __restrict__ P1,
                        const float* __restrict__ P2,
                        float* ws, float* out) {
    int t = threadIdx.x, lane = t & 31, w = t >> 5;
    double aent = 0, ach = 0, acorr = 0, acos_ = 0;
    const float* pose = ws + O_POSE;
    for (int i = t; i < kBN; i += 256) {
        int b = i >> 10;
        aent += (double)ws[O_ROWENT1 + i] + (double)ws[O_ROWENT2 + i];
        {
            float S1 = ws[O_CS1_1 + i], S2 = ws[O_CS2_1 + i];
            float cm = ws[O_COLMAX1 + i], cs = ws[O_COLSUM1 + i];
            aent += (double)(-S2 + (cm + __logf(cs)) * S1);
            S1 = ws[O_CS1_2 + i]; S2 = ws[O_CS2_2 + i];
            cm = ws[O_COLMAX2 + i]; cs = ws[O_COLSUM2 + i];
            aent += (double)(-S2 + (cm + __logf(cs)) * S1);
        }
        ach += (double)ws[O_RMIN0 + i] + (double)ws[O_CMIN0 + i]
             + (double)ws[O_RMIN1 + i] + (double)ws[O_CMIN1 + i];
        const float* ps = pose + b * 24;
        // corr(p1in2, p1to2)
        {
            float p0 = P1[3 * i], p1 = P1[3 * i + 1], p2 = P1[3 * i + 2];
            float tx = fmaf(ps[0], p0, fmaf(ps[1], p1, fmaf(ps[2], p2, ps[9])));
            float ty = fmaf(ps[3], p0, fmaf(ps[4], p1, fmaf(ps[5], p2, ps[10])));
            float tz = fmaf(ps[6], p0, fmaf(ps[7], p1, fmaf(ps[8], p2, ps[11])));
            float d0 = fabsf(ws[O_P1IN2 + 3 * i] - tx);
            float d1 = fabsf(ws[O_P1IN2 + 3 * i + 1] - ty);
            float d2 = fabsf(ws[O_P1IN2 + 3 * i + 2] - tz);
            acorr += (double)((d0 > 0.1f ? d0 - 0.05f : d0 * d0 * 5.0f)
                            + (d1 > 0.1f ? d1 - 0.05f : d1 * d1 * 5.0f)
                            + (d2 > 0.1f ? d2 - 0.05f : d2 * d2 * 5.0f));
        }
        // corr(p2in1, p2to1)
        {
            const float* qs = ps + 12;
            float p0 = P2[3 * i], p1 = P2[3 * i + 1], p2 = P2[3 * i + 2];
            float tx = fmaf(qs[0], p0, fmaf(qs[1], p1, fmaf(qs[2], p2, qs[9])));
            float ty = fmaf(qs[3], p0, fmaf(qs[4], p1, fmaf(qs[5], p2, qs[10])));
            float tz = fmaf(qs[6], p0, fmaf(qs[7], p1, fmaf(qs[8], p2, qs[11])));
            float d0 = fabsf(ws[O_P2IN1 + 3 * i] - tx);
            float d1 = fabsf(ws[O_P2IN1 + 3 * i + 1] - ty);
            float d2 = fabsf(ws[O_P2IN1 + 3 * i + 2] - tz);
            acorr += (double)((d0 > 0.1f ? d0 - 0.05f : d0 * d0 * 5.0f)
                            + (d1 > 0.1f ? d1 - 0.05f : d1 * d1 * 5.0f)
                            + (d2 > 0.1f ? d2 - 0.05f : d2 * d2 * 5.0f));
        }
    }
    for (int b = t; b < kB; b += 256) {
        float d = ws[O_COS + b], s1 = ws[O_COS + 32 + b], s2 = ws[O_COS + 64 + b];
        float na = fmaxf(sqrtf(s1), 1e-8f), nb = fmaxf(sqrtf(s2), 1e-8f);
        acos_ += (double)(1.0f - d / (na * nb));
    }
    double part = ach / (double)kBN + acorr * 1.0 / (double)kBN
                + aent * 0.0001 / (double)kBN + acos_ / (double)kB;
#pragma unroll
    for (int o = 16; o > 0; o >>= 1) part += __shfl_xor(part, o, 32);
    __shared__ double red[8];
    if (lane == 0) red[w] = part;
    __syncthreads();
    if (t == 0) {
        double s = 0;
        for (int i = 0; i < 8; ++i) s += red[i];
        out[0] = (float)s;
    }
}

// ---------------- host launcher ----------------
extern "C" void kernel_launch(void* const* d_in, const int* in_sizes, int n_in,
                              void* d_out, int out_size, void* d_ws, size_t ws_size,
                              hipStream_t stream) {
    (void)in_sizes; (void)n_in; (void)out_size; (void)ws_size;
    const float* P1    = (const float*)d_in[0];
    const float* P2    = (const float*)d_in[1];
    const float* A1    = (const float*)d_in[2];
    const float* A2    = (const float*)d_in[3];
    const float* scale = (const float*)d_in[4];
    const float* rot   = (const float*)d_in[5];
    const float* trans = (const float*)d_in[6];
    float* ws  = (float*)d_ws;
    float* out = (float*)d_out;

    int initBlocks = (int)((WS_FLOATS + 255) / 256);
    k_init<<<initBlocks, 256, 0, stream>>>(ws);
    k_pose<<<1, 32, 0, stream>>>(scale, rot, trans, ws);
    k_pass1<<<dim3(kB, 4, 4), 256, 0, stream>>>(A1, A2, ws);
    k_softmax<<<dim3(kB, kN / 8), 256, 0, stream>>>(A1, P2, ws, 0);
    k_softmax<<<dim3(kB, kN / 8), 256, 0, stream>>>(A2, P1, ws, 1);
    k_chamfer<<<dim3(kB, 2, 8), 256, 0, stream>>>(P1, P2, ws);
    k_final<<<1, 256, 0, stream>>>(P1, P2, ws, out);
}